// NNAttention_59081570124649
// MI455X (gfx1250) — compile-verified
//
#include <hip/hip_runtime.h>

#define Bn 32
#define Tn 4096
#define Cn 256
#define Kn 15
#define TT 64        // t-tile per block in score kernel
#define CS 260       // conv tile LDS stride (mult of 4 for float4 align, %64==4 kills bank conflicts)

typedef __attribute__((ext_vector_type(2))) float v2f;
typedef __attribute__((ext_vector_type(8))) float v8f;

// ---------------------------------------------------------------------------
// Kernel 1: score[b,t] = sum_c relu(eh + dhx + conv_b + conv) * lin_w + lin_b
// conv tile (256 x 64) computed as WMMA GEMM: W(Cx16, k15 zero-padded) x Toeplitz(ax_prev)
// ---------------------------------------------------------------------------
__global__ __launch_bounds__(256) void nna_score_kernel(
    const float* __restrict__ eh, const float* __restrict__ dhx,
    const float* __restrict__ ax_prev, const float* __restrict__ conv_w,
    const float* __restrict__ conv_b, const float* __restrict__ lin_w,
    const float* __restrict__ lin_b, float* __restrict__ score_out)
{
  __shared__ __align__(16) float convt[TT * CS];   // [t_local][c], stride CS
  __shared__ __align__(16) float base_c[Cn];       // dhx + conv_b
  __shared__ __align__(16) float linw[Cn];
  __shared__ __align__(16) float win[TT + 32];     // ax_prev sliding window

  const int tid  = threadIdx.x;
  const int lane = tid & 31;
  const int wv   = tid >> 5;      // 0..7
  const int ln   = lane & 15;     // lane within half-wave
  const int hl   = lane >> 4;     // half-wave select
  const int b    = blockIdx.x;
  const int t0   = blockIdx.y * TT;

  base_c[tid] = dhx[b * Cn + tid] + conv_b[tid];
  linw[tid]   = lin_w[tid];
  if (tid < TT + 16) {
    int tg = t0 - 7 + tid;        // win[i] = ax_prev[b, t0-7+i], zero OOB
    win[tid] = (tg >= 0 && tg < Tn) ? ax_prev[b * Tn + tg] : 0.0f;
  }
  __syncthreads();

  // A tiles (weights) in registers: wave covers c-strips {wv, wv+8}, K 15->16 padded.
  // A 16x4 f32 layout: lanes 0-15 M=0..15, VGPR0 K={0|2 by half}, VGPR1 K={1|3}.
  float a0[2][4], a1[2][4];
  #pragma unroll
  for (int si = 0; si < 2; ++si) {
    const int c = (wv + 8 * si) * 16 + ln;
    #pragma unroll
    for (int q = 0; q < 4; ++q) {
      const int k0 = 4 * q + 2 * hl;
      a0[si][q] = (k0 < Kn)     ? conv_w[c * Kn + k0]     : 0.0f;
      a1[si][q] = (k0 + 1 < Kn) ? conv_w[c * Kn + k0 + 1] : 0.0f;
    }
  }

  // conv[c, t_local] = sum_k w[c,k] * win[t_local + k]
  #pragma unroll
  for (int jt = 0; jt < 4; ++jt) {          // 16-t sub-tiles
    #pragma unroll
    for (int si = 0; si < 2; ++si) {        // this wave's two c-strips
      v8f acc = {0.f,0.f,0.f,0.f,0.f,0.f,0.f,0.f};
      #pragma unroll
      for (int q = 0; q < 4; ++q) {         // K chunks of 4
        v2f av, bv;
        av.x = a0[si][q]; av.y = a1[si][q];
        const int widx = 16 * jt + ln + 4 * q + 2 * hl;  // B[k][n] = win[16jt + n + k]
        bv.x = win[widx]; bv.y = win[widx + 1];
        acc = __builtin_amdgcn_wmma_f32_16x16x4_f32(
            false, av, false, bv, (short)0, acc, false, false);
      }
      // D 16x16: VGPR j -> M = j + 8*hl (c), N = ln (t)
      const int tl = 16 * jt + ln;
      const int cb = (wv + 8 * si) * 16 + 8 * hl;
      #pragma unroll
      for (int j = 0; j < 8; ++j)
        convt[tl * CS + cb + j] = acc[j];
    }
  }
  __syncthreads();

  // Streaming phase: relu-dot over c, coalesced float4 eh reads.
  const float lb = lin_b[0];
  #pragma unroll
  for (int i = 0; i < 8; ++i) {
    const int tl = wv * 8 + i;
    const long gbase = ((long)(b * Tn + t0 + tl)) * Cn;
    float accs = 0.f;
    #pragma unroll
    for (int ch = 0; ch < 2; ++ch) {
      const int c = ch * 128 + lane * 4;
      const float4 e  = *(const float4*)(eh + gbase + c);
      const float4 cv = *(const float4*)(convt + tl * CS + c);
      const float4 bs = *(const float4*)(base_c + c);
      const float4 lw = *(const float4*)(linw + c);
      const float p0 = fmaxf(e.x + cv.x + bs.x, 0.f);
      const float p1 = fmaxf(e.y + cv.y + bs.y, 0.f);
      const float p2 = fmaxf(e.z + cv.z + bs.z, 0.f);
      const float p3 = fmaxf(e.w + cv.w + bs.w, 0.f);
      accs += p0 * lw.x + p1 * lw.y + p2 * lw.z + p3 * lw.w;
    }
    #pragma unroll
    for (int off = 16; off > 0; off >>= 1)
      accs += __shfl_xor(accs, off, 32);
    if (lane == 0) score_out[b * Tn + t0 + tl] = accs + lb;
  }
}

// ---------------------------------------------------------------------------
// Kernel 2: in-place softmax over T per batch row (row kept in registers)
// ---------------------------------------------------------------------------
__global__ __launch_bounds__(256) void nna_softmax_kernel(float* __restrict__ sa)
{
  __shared__ float red[8];
  const int b = blockIdx.x;
  const int tid = threadIdx.x;
  const int lane = tid & 31;
  const int wv = tid >> 5;

  float v[16];
  float m = -3.402823466e38f;
  #pragma unroll
  for (int i = 0; i < 16; ++i) {
    v[i] = sa[b * Tn + tid + 256 * i];
    m = fmaxf(m, v[i]);
  }
  #pragma unroll
  for (int off = 16; off > 0; off >>= 1)
    m = fmaxf(m, __shfl_xor(m, off, 32));
  if (lane == 0) red[wv] = m;
  __syncthreads();
  m = red[0];
  #pragma unroll
  for (int w = 1; w < 8; ++w) m = fmaxf(m, red[w]);
  __syncthreads();

  float s = 0.f;
  #pragma unroll
  for (int i = 0; i < 16; ++i) { v[i] = __expf(v[i] - m); s += v[i]; }
  #pragma unroll
  for (int off = 16; off > 0; off >>= 1)
    s += __shfl_xor(s, off, 32);
  if (lane == 0) red[wv] = s;
  __syncthreads();
  s = 0.f;
  #pragma unroll
  for (int w = 0; w < 8; ++w) s += red[w];

  const float inv = 1.0f / s;
  #pragma unroll
  for (int i = 0; i < 16; ++i) sa[b * Tn + tid + 256 * i] = v[i] * inv;
}

// ---------------------------------------------------------------------------
// Kernel 3: sx[b,c] = sum_t eh[b,t,c] * ax[b,t]  via WMMA f32 16x16x4
// A[m][k] = eh[t0+k, c0+m] (lanes = c, coalesced 64B segments),
// B[k][n] = ax[t0+k] replicated across N (value depends only on K -> layout-robust)
// ---------------------------------------------------------------------------
__global__ __launch_bounds__(256) void nna_sx_kernel(
    const float* __restrict__ eh, const float* __restrict__ ax,
    float* __restrict__ sx)
{
  __shared__ float axs[Tn];          // 16 KB staged ax row
  __shared__ float red[8][16];
  const int b    = blockIdx.x;
  const int cg   = blockIdx.y;       // 64-c group, 0..3
  const int tid  = threadIdx.x;
  const int lane = tid & 31;
  const int wv   = tid >> 5;
  const int ln   = lane & 15;
  const int hl   = lane >> 4;

  for (int i = tid; i < Tn; i += 256) axs[i] = ax[b * Tn + i];
  __syncthreads();

  const int strip  = wv & 3;          // 16-c strip within group
  const int th     = wv >> 2;         // t-range half
  const int c0     = cg * 64 + strip * 16;
  const int tstart = th * 2048;
  const float* ep  = eh + ((long)b * Tn + tstart) * Cn + c0 + ln;

  v8f acc4[4];
  #pragma unroll
  for (int u = 0; u < 4; ++u) acc4[u] = (v8f){0.f,0.f,0.f,0.f,0.f,0.f,0.f,0.f};

  int tt = tstart;
  for (int stepo = 0; stepo < 128; ++stepo) {     // 128 x 4 = 512 k-chunks of 4
    #pragma unroll
    for (int u = 0; u < 4; ++u) {                 // 4 independent WMMA chains
      v2f av, bv;
      av.x = ep[(2 * hl) * Cn];
      av.y = ep[(2 * hl + 1) * Cn];
      bv.x = axs[tt + 2 * hl];
      bv.y = axs[tt + 2 * hl + 1];
      acc4[u] = __builtin_amdgcn_wmma_f32_16x16x4_f32(
          false, av, false, bv, (short)0, acc4[u], false, false);
      ep += 4 * Cn;
      tt += 4;
    }
  }
  const v8f acc = (acc4[0] + acc4[1]) + (acc4[2] + acc4[3]);

  // D columns are replicated; take N==0 column: lanes 0 (M=j) and 16 (M=8+j)
  if (ln == 0) {
    #pragma unroll
    for (int j = 0; j < 8; ++j) red[wv][8 * hl + j] = acc[j];
  }
  __syncthreads();
  if (tid < 64) {
    const int s = tid >> 4, cc = tid & 15;
    sx[b * Cn + cg * 64 + tid] = red[s][cc] + red[s + 4][cc];
  }
}

// ---------------------------------------------------------------------------
extern "C" void kernel_launch(void* const* d_in, const int* in_sizes, int n_in,
                              void* d_out, int out_size, void* d_ws, size_t ws_size,
                              hipStream_t stream) {
  const float* eh      = (const float*)d_in[0];
  const float* dhx     = (const float*)d_in[1];
  const float* ax_prev = (const float*)d_in[2];
  const float* conv_w  = (const float*)d_in[3];
  const float* conv_b  = (const float*)d_in[4];
  const float* lin_w   = (const float*)d_in[5];
  const float* lin_b   = (const float*)d_in[6];

  float* out = (float*)d_out;
  float* sx  = out;                 // (B,1,C) = 8192 floats
  float* axo = out + Bn * Cn;       // (B,T)   = 131072 floats; also holds raw scores

  dim3 g1(Bn, Tn / TT);             // 32 x 64 blocks
  nna_score_kernel<<<g1, 256, 0, stream>>>(eh, dhx, ax_prev, conv_w, conv_b,
                                           lin_w, lin_b, axo);
  nna_softmax_kernel<<<Bn, 256, 0, stream>>>(axo);
  dim3 g3(Bn, Cn / 64);             // 32 x 4 blocks
  nna_sx_kernel<<<g3, 256, 0, stream>>>(eh, axo, sx);
}